// ReadoutBlock_50190987821195
// MI455X (gfx1250) — compile-verified
//
#include <hip/hip_runtime.h>
#include <hip/hip_bf16.h>
#include <math.h>

typedef float v2f __attribute__((ext_vector_type(2)));
typedef float v8f __attribute__((ext_vector_type(8)));

#define N_NODES  500000
#define G_GRAPHS 1024
#define D_CH     128          // in_channels == out_channels
#define K_TOT    (2 * D_CH)   // 256

__device__ __forceinline__ float sigmoidf_(float v) {
    return 1.0f / (1.0f + __expf(-v));
}

// ---------------------------------------------------------------------------
// Kernel 0: zero workspace accumulators + the loss slot of d_out
// ---------------------------------------------------------------------------
__global__ void zero_kernel(float* __restrict__ gf_accum,
                            float* __restrict__ seg_sum,
                            float* __restrict__ counts,
                            float* __restrict__ out_loss) {
    const int i = blockIdx.x * blockDim.x + threadIdx.x;
    if (i < G_GRAPHS * D_CH) gf_accum[i] = 0.0f;
    if (i < G_GRAPHS) { seg_sum[i] = 0.0f; counts[i] = 0.0f; }
    if (i == 0) out_loss[0] = 0.0f;
}

// ---------------------------------------------------------------------------
// Kernel 1: fused node GEMM (sigmoid MLP gate) + weighted segment pooling
//           + per-node gate-norm partial reductions.
// One wave handles 32 nodes (two 16x16 M-tiles sharing B), all 128 output
// columns, K=256 via 64 x V_WMMA_F32_16X16X4_F32 steps per M-tile pair.
// ---------------------------------------------------------------------------
__global__ __launch_bounds__(256) void node_gemm_pool_kernel(
    const float* __restrict__ x,        // [N, D]
    const int*   __restrict__ batch,    // [N]
    const float* __restrict__ ginfo,    // [G, D]
    const float* __restrict__ mlp_w,    // [D, 2D] row-major
    const float* __restrict__ mlp_b,    // [D]
    float* __restrict__ gf_accum,       // [G, D] (pre-zeroed)
    float* __restrict__ seg_sum,        // [G]
    float* __restrict__ counts)         // [G]
{
    const int wave_in_block = threadIdx.x >> 5;
    const int tile = blockIdx.x * 8 + wave_in_block;   // 32-node tile index
    const int numTiles = N_NODES / 32;                  // 15625, exact
    if (tile >= numTiles) return;                       // wave-uniform: EXEC stays all-ones

    const int l    = threadIdx.x & 31;
    const int lsub = l & 15;      // N-col / M-row sub-lane
    const int lhi  = l >> 4;      // selects K pair (A) / row-half (C)
    const int base = tile * 32;

    // ---- A-matrix per-lane source pointers (WMMA 16x4 f32 layout) ----
    // lane holds M=lsub, K pair {k0+2*lhi, k0+2*lhi+1}
    const int nodeA0 = base + lsub;            // M-tile 0
    const int nodeA1 = base + 16 + lsub;       // M-tile 1
    const float* pA0 = x + (size_t)nodeA0 * D_CH + 2 * lhi;
    const float* pA1 = x + (size_t)nodeA1 * D_CH + 2 * lhi;
    const float* pG0 = ginfo + (size_t)batch[nodeA0] * D_CH + 2 * lhi;
    const float* pG1 = ginfo + (size_t)batch[nodeA1] * D_CH + 2 * lhi;
    // ---- B-matrix per-lane pointer: B[k][n] = mlp_w[n*256 + k]; a lane's
    // (v0,v1) pair is two consecutive floats of row n = n0+lsub ----
    const float* pB = mlp_w + (size_t)lsub * K_TOT + 2 * lhi;

    v8f zero8;
    #pragma unroll
    for (int i = 0; i < 8; ++i) zero8[i] = 0.0f;

    v8f acc[2][8];   // [M-tile][N-tile] accumulators: 128 VGPRs
    #pragma unroll
    for (int mt = 0; mt < 2; ++mt)
        #pragma unroll
        for (int n = 0; n < 8; ++n) acc[mt][n] = zero8;

    // ---- K = 0..127 : cat half from x ----
    #pragma unroll 4
    for (int kk = 0; kk < 32; ++kk) {
        const int k0 = kk * 4;
        const v2f a0 = *(const v2f*)(pA0 + k0);
        const v2f a1 = *(const v2f*)(pA1 + k0);
        #pragma unroll
        for (int n = 0; n < 8; ++n) {
            const v2f b = *(const v2f*)(pB + n * 16 * K_TOT + k0);
            acc[0][n] = __builtin_amdgcn_wmma_f32_16x16x4_f32(
                false, a0, false, b, (short)0, acc[0][n], false, false);
            acc[1][n] = __builtin_amdgcn_wmma_f32_16x16x4_f32(
                false, a1, false, b, (short)0, acc[1][n], false, false);
        }
    }
    // ---- K = 128..255 : cat half from gathered global_information ----
    #pragma unroll 4
    for (int kk = 0; kk < 32; ++kk) {
        const int k0 = kk * 4;
        const v2f a0 = *(const v2f*)(pG0 + k0);
        const v2f a1 = *(const v2f*)(pG1 + k0);
        #pragma unroll
        for (int n = 0; n < 8; ++n) {
            const v2f b = *(const v2f*)(pB + n * 16 * K_TOT + D_CH + k0);
            acc[0][n] = __builtin_amdgcn_wmma_f32_16x16x4_f32(
                false, a0, false, b, (short)0, acc[0][n], false, false);
            acc[1][n] = __builtin_amdgcn_wmma_f32_16x16x4_f32(
                false, a1, false, b, (short)0, acc[1][n], false, false);
        }
    }

    // ---- epilogue: sigmoid, weighted pool atomics, gate-norm partials ----
    // C layout: vgpr r, lanes 0-15 -> row r, lanes 16-31 -> row r+8; col = lsub+16n
    int bnode[2][8];
    #pragma unroll
    for (int mt = 0; mt < 2; ++mt)
        #pragma unroll
        for (int r = 0; r < 8; ++r)
            bnode[mt][r] = batch[base + mt * 16 + r + 8 * lhi];

    float sum2[2][8];
    #pragma unroll
    for (int mt = 0; mt < 2; ++mt)
        #pragma unroll
        for (int r = 0; r < 8; ++r) sum2[mt][r] = 0.0f;

    #pragma unroll
    for (int mt = 0; mt < 2; ++mt) {
        #pragma unroll
        for (int n = 0; n < 8; ++n) {
            const int col = n * 16 + lsub;
            const float bias = mlp_b[col];
            #pragma unroll
            for (int r = 0; r < 8; ++r) {
                const float w = sigmoidf_(acc[mt][n][r] + bias);
                const int node = base + mt * 16 + r + 8 * lhi;
                const float xv = x[(size_t)node * D_CH + col];   // coalesced, L2-hot
                atomicAdd(&gf_accum[bnode[mt][r] * D_CH + col], w * xv);
                sum2[mt][r] += w * w;
            }
        }
    }

    // reduce ||w||^2 across the 16 lanes holding one row, emit norm atomics
    #pragma unroll
    for (int mt = 0; mt < 2; ++mt) {
        #pragma unroll
        for (int r = 0; r < 8; ++r) {
            float s = sum2[mt][r];
            s += __shfl_xor(s, 1, 32);
            s += __shfl_xor(s, 2, 32);
            s += __shfl_xor(s, 4, 32);
            s += __shfl_xor(s, 8, 32);
            if (lsub == 0) {   // lane 0 -> row r ; lane 16 -> row r+8
                const int b = bnode[mt][r];
                atomicAdd(&seg_sum[b], sqrtf(s));
                atomicAdd(&counts[b], 1.0f);
            }
        }
    }
}

// ---------------------------------------------------------------------------
// Kernel 2: per-graph GRUCell + gate-loss finalization (tiny: 0.2 GFLOP)
// One block per graph; thread j computes gate rows gi[j], gh[j].
// ---------------------------------------------------------------------------
__global__ __launch_bounds__(384) void gru_loss_kernel(
    const float* __restrict__ gf_new,   // [G, D] pooled accumulator
    const float* __restrict__ ginfo,    // [G, D] hidden state h
    const float* __restrict__ w_ih,     // [3D, D]
    const float* __restrict__ w_hh,     // [3D, D]
    const float* __restrict__ b_ih,     // [3D]
    const float* __restrict__ b_hh,     // [3D]
    const float* __restrict__ seg_sum,  // [G]
    const float* __restrict__ counts,   // [G]
    float* __restrict__ out)            // [G*D + 1]
{
    __shared__ float s_a[D_CH], s_h[D_CH], s_gi[3 * D_CH], s_gh[3 * D_CH];
    const int g = blockIdx.x;
    const int t = threadIdx.x;

    if (t < D_CH)            s_a[t] = gf_new[g * D_CH + t];
    else if (t < 2 * D_CH)   s_h[t - D_CH] = ginfo[g * D_CH + (t - D_CH)];
    __syncthreads();

    float accI = b_ih[t];
    float accH = b_hh[t];
    const float* wi = w_ih + (size_t)t * D_CH;
    const float* wh = w_hh + (size_t)t * D_CH;
    #pragma unroll 8
    for (int k = 0; k < D_CH; ++k) {
        accI = fmaf(wi[k], s_a[k], accI);
        accH = fmaf(wh[k], s_h[k], accH);
    }
    s_gi[t] = accI;
    s_gh[t] = accH;
    __syncthreads();

    if (t < D_CH) {
        const float r  = sigmoidf_(s_gi[t] + s_gh[t]);
        const float z  = sigmoidf_(s_gi[D_CH + t] + s_gh[D_CH + t]);
        const float nn = tanhf(s_gi[2 * D_CH + t] + r * s_gh[2 * D_CH + t]);
        out[g * D_CH + t] = (1.0f - z) * nn + z * s_h[t];
    }
    if (t == 0) {
        const float term = seg_sum[g] / fmaxf(counts[g], 1.0f);
        atomicAdd(&out[G_GRAPHS * D_CH], term * (1.0f / (float)G_GRAPHS));
    }
}

// ---------------------------------------------------------------------------
extern "C" void kernel_launch(void* const* d_in, const int* in_sizes, int n_in,
                              void* d_out, int out_size, void* d_ws, size_t ws_size,
                              hipStream_t stream) {
    const float* x      = (const float*)d_in[0];
    const int*   batch  = (const int*)  d_in[1];
    const float* ginfo  = (const float*)d_in[2];
    const float* mlp_w  = (const float*)d_in[3];
    const float* mlp_b  = (const float*)d_in[4];
    const float* w_ih   = (const float*)d_in[5];
    const float* w_hh   = (const float*)d_in[6];
    const float* b_ih   = (const float*)d_in[7];
    const float* b_hh   = (const float*)d_in[8];
    float* out = (float*)d_out;

    float* gf_accum = (float*)d_ws;                       // G*D floats
    float* seg_sum  = gf_accum + G_GRAPHS * D_CH;         // G floats
    float* counts   = seg_sum + G_GRAPHS;                 // G floats

    zero_kernel<<<(G_GRAPHS * D_CH + 255) / 256, 256, 0, stream>>>(
        gf_accum, seg_sum, counts, out + G_GRAPHS * D_CH);

    const int numTiles = N_NODES / 32;                    // 15625 (exact)
    const int blocks = (numTiles + 7) / 8;                // 8 waves / block
    node_gemm_pool_kernel<<<blocks, 256, 0, stream>>>(
        x, batch, ginfo, mlp_w, mlp_b, gf_accum, seg_sum, counts);

    gru_loss_kernel<<<G_GRAPHS, 384, 0, stream>>>(
        gf_accum, ginfo, w_ih, w_hh, b_ih, b_hh, seg_sum, counts, out);
}